// AttnConv_61332132987151
// MI455X (gfx1250) — compile-verified
//
#include <hip/hip_runtime.h>

typedef float v2f __attribute__((ext_vector_type(2)));
typedef float v8f __attribute__((ext_vector_type(8)));

#define H      128
#define PSTR   272        // P row stride / W2 column count: 128 (src proj) + 128 (dst proj) + 2 scores + pad
#define EPS    1e-5f

// ---------------- zero a float span ----------------
__global__ void k_zero(float* __restrict__ p, long n) {
  long i = (long)blockIdx.x * blockDim.x + threadIdx.x;
  long st = (long)gridDim.x * blockDim.x;
  for (; i < n; i += st) p[i] = 0.f;
}

// ---------------- build fused weight matrix W2 (128 x PSTR) ----------------
// col 0..127   : W_top_g  (Wg[h][r][d],      c = h*32+d)
// col 128..255 : W_bot_g  (Wg[h][r+128][d])
// col 256      : W_f[r]   (top half)
// col 257      : W_f[r+128] (bottom half)
// col 258..271 : zero pad
__global__ void k_build_w2(const float* __restrict__ W_f, const float* __restrict__ Wg,
                           float* __restrict__ W2) {
  int t = blockIdx.x * blockDim.x + threadIdx.x;
  if (t >= H * PSTR) return;
  int row = t / PSTR, col = t % PSTR;
  float v = 0.f;
  if (col < 128) {
    int h = col >> 5, d = col & 31;
    v = Wg[h * (256 * 32) + row * 32 + d];
  } else if (col < 256) {
    int c2 = col - 128, h = c2 >> 5, d = c2 & 31;
    v = Wg[h * (256 * 32) + (row + 128) * 32 + d];
  } else if (col == 256) {
    v = W_f[row];
  } else if (col == 257) {
    v = W_f[row + 128];
  }
  W2[t] = v;
}

// ---------------- WMMA f32 GEMM: P(N x PSTR) = X(N x 128) @ W2(128 x PSTR) ----------------
// One wave computes one 16x16 output tile via V_WMMA_F32_16X16X4_F32, K-loop of 32 steps.
// A 16x4 layout: lanes 0-15 hold (M=lane, K=k,k+1) in v0,v1; lanes 16-31 hold K=k+2,k+3.
// B 4x16 layout: lanes 0-15 hold (K=k,  N=lane) in v0 / (K=k+1) in v1; lanes 16-31 K=k+2/k+3.
__global__ void k_gemm_wmma(const float* __restrict__ X, const float* __restrict__ W2,
                            float* __restrict__ P) {
  int wave  = threadIdx.x >> 5;
  int lane  = threadIdx.x & 31;
  int ntile = blockIdx.y * 4 + wave;
  if (ntile >= PSTR / 16) return;
  int m0   = blockIdx.x << 4;
  int n0   = ntile << 4;
  int mrow = m0 + (lane & 15);
  int koff = (lane >> 4) << 1;            // 0 for lanes 0-15, 2 for lanes 16-31
  int ncol = n0 + (lane & 15);
  const float* xr = X + (long)mrow * H + koff;
  const float* wr = W2 + (long)koff * PSTR + ncol;
  v8f c = {};
  for (int k = 0; k < H; k += 4) {
    v2f a, b;
    a.x = xr[k];
    a.y = xr[k + 1];
    b.x = wr[k * PSTR];
    b.y = wr[(k + 1) * PSTR];
    c = __builtin_amdgcn_wmma_f32_16x16x4_f32(false, a, false, b, (short)0, c, false, false);
  }
  int mbase = m0 + ((lane >> 4) << 3);    // rows r..r+7 live in lanes 0-15, r+8..r+15 in 16-31
#pragma unroll
  for (int r = 0; r < 8; ++r)
    P[(long)(mbase + r) * PSTR + ncol] = c[r];
}

// ---------------- edge BN statistics (sum / sumsq per g-channel + f scalar) ----------------
__global__ void k_edge_stats(const float* __restrict__ P, const int* __restrict__ src,
                             const int* __restrict__ dst, float* __restrict__ estats, int E) {
  __shared__ float ls[258];
  for (int i = threadIdx.x; i < 258; i += blockDim.x) ls[i] = 0.f;
  __syncthreads();
  int lane   = threadIdx.x & 31;
  int wgl    = blockIdx.x * (blockDim.x >> 5) + (threadIdx.x >> 5);
  int nwaves = gridDim.x * (blockDim.x >> 5);
  float s0 = 0, s1 = 0, s2 = 0, s3 = 0, q0 = 0, q1 = 0, q2 = 0, q3 = 0, fs = 0, fq = 0;
  for (int e = wgl; e < E; e += nwaves) {
    int s = src[e], d = dst[e];
    const float* ps = P + (long)s * PSTR;
    const float* pd = P + (long)d * PSTR + H;
    float v0 = ps[lane]      + pd[lane];
    float v1 = ps[lane + 32] + pd[lane + 32];
    float v2 = ps[lane + 64] + pd[lane + 64];
    float v3 = ps[lane + 96] + pd[lane + 96];
    s0 += v0; q0 += v0 * v0;
    s1 += v1; q1 += v1 * v1;
    s2 += v2; q2 += v2 * v2;
    s3 += v3; q3 += v3 * v3;
    if (lane == 0) {
      float sv = ps[256] + pd[129];       // P[s][256] + P[d][257]
      fs += sv; fq += sv * sv;
    }
  }
  atomicAdd(&ls[lane],       s0); atomicAdd(&ls[128 + lane],      q0);
  atomicAdd(&ls[lane + 32],  s1); atomicAdd(&ls[128 + lane + 32], q1);
  atomicAdd(&ls[lane + 64],  s2); atomicAdd(&ls[128 + lane + 64], q2);
  atomicAdd(&ls[lane + 96],  s3); atomicAdd(&ls[128 + lane + 96], q3);
  if (lane == 0) { atomicAdd(&ls[256], fs); atomicAdd(&ls[257], fq); }
  __syncthreads();
  for (int i = threadIdx.x; i < 258; i += blockDim.x) atomicAdd(&estats[i], ls[i]);
}

// ---------------- finalize edge BN params: mean + invstd ----------------
__global__ void k_fin_edge(const float* __restrict__ estats, float* __restrict__ eparams, int E) {
  int c = threadIdx.x;
  float invE = 1.f / (float)E;
  if (c < 128) {
    float m = estats[c] * invE;
    float v = estats[128 + c] * invE - m * m;
    eparams[c] = m;
    eparams[128 + c] = rsqrtf(v + EPS);
  } else if (c == 128) {
    float m = estats[256] * invE;
    float v = estats[257] * invE - m * m;
    eparams[256] = m;
    eparams[257] = rsqrtf(v + EPS);
  }
}

// ---------------- per-edge exp score + segment-sum denominators ----------------
__global__ void k_edge_exp(const float* __restrict__ P, const int* __restrict__ src,
                           const int* __restrict__ dst, const float* __restrict__ eparams,
                           const float* __restrict__ gamma_f, const float* __restrict__ beta_f,
                           float* __restrict__ exp_arr, float* __restrict__ sum_exp, int E) {
  int e = blockIdx.x * blockDim.x + threadIdx.x;
  if (e >= E) return;
  int s = src[e], d = dst[e];
  float sv = P[(long)s * PSTR + 256] + P[(long)d * PSTR + 257];
  float z  = gamma_f[0] * (sv - eparams[256]) * eparams[257] + beta_f[0];
  z = fmaxf(z, 0.f);
  float ee = __expf(z);
  exp_arr[e] = ee;
  atomicAdd(&sum_exp[d], ee);
}

// ---------------- per-edge normalize/ReLU/weight + scatter-add ----------------
__global__ void k_edge_scatter(const float* __restrict__ P, const int* __restrict__ src,
                               const int* __restrict__ dst, const float* __restrict__ eparams,
                               const float* __restrict__ gamma_g, const float* __restrict__ beta_g,
                               const float* __restrict__ exp_arr, const float* __restrict__ sum_exp,
                               float* __restrict__ hacc, int E) {
  int lane   = threadIdx.x & 31;
  int wgl    = blockIdx.x * (blockDim.x >> 5) + (threadIdx.x >> 5);
  int nwaves = gridDim.x * (blockDim.x >> 5);
  float m0 = eparams[lane],      i0 = eparams[128 + lane];
  float m1 = eparams[lane + 32], i1 = eparams[160 + lane];
  float m2 = eparams[lane + 64], i2 = eparams[192 + lane];
  float m3 = eparams[lane + 96], i3 = eparams[224 + lane];
  float g0 = gamma_g[lane],      b0 = beta_g[lane];
  float g1 = gamma_g[lane + 32], b1 = beta_g[lane + 32];
  float g2 = gamma_g[lane + 64], b2 = beta_g[lane + 64];
  float g3 = gamma_g[lane + 96], b3 = beta_g[lane + 96];
  for (int e = wgl; e < E; e += nwaves) {
    int s = src[e], d = dst[e];
    const float* ps = P + (long)s * PSTR;
    const float* pd = P + (long)d * PSTR + H;
    float a = exp_arr[e] / sum_exp[d];
    float* hd = hacc + (long)d * H;
    float v;
    v = ps[lane]      + pd[lane];      v = fmaxf(g0 * (v - m0) * i0 + b0, 0.f); atomicAdd(&hd[lane],      a * v);
    v = ps[lane + 32] + pd[lane + 32]; v = fmaxf(g1 * (v - m1) * i1 + b1, 0.f); atomicAdd(&hd[lane + 32], a * v);
    v = ps[lane + 64] + pd[lane + 64]; v = fmaxf(g2 * (v - m2) * i2 + b2, 0.f); atomicAdd(&hd[lane + 64], a * v);
    v = ps[lane + 96] + pd[lane + 96]; v = fmaxf(g3 * (v - m3) * i3 + b3, 0.f); atomicAdd(&hd[lane + 96], a * v);
  }
}

// ---------------- node BN statistics ----------------
__global__ void k_node_stats(const float* __restrict__ hacc, float* __restrict__ nstats, int N) {
  int c = threadIdx.x;                   // 0..127 = channel
  int rows = (N + gridDim.x - 1) / gridDim.x;
  int r0 = blockIdx.x * rows;
  int r1 = min(r0 + rows, N);
  float s = 0.f, q = 0.f;
  for (int n = r0; n < r1; ++n) {
    float v = hacc[(long)n * H + c];
    s += v; q += v * v;
  }
  atomicAdd(&nstats[c], s);
  atomicAdd(&nstats[128 + c], q);
}

__global__ void k_fin_node(const float* __restrict__ nstats, float* __restrict__ nparams, int N) {
  int c = threadIdx.x;
  if (c >= 128) return;
  float invN = 1.f / (float)N;
  float m = nstats[c] * invN;
  float v = nstats[128 + c] * invN - m * m;
  nparams[c] = m;
  nparams[128 + c] = rsqrtf(v + EPS);
}

// ---------------- output: node BN + residual ----------------
__global__ void k_out(const float* __restrict__ hacc, const float* __restrict__ nparams,
                      const float* __restrict__ gamma_n, const float* __restrict__ beta_n,
                      const float* __restrict__ x_in, float* __restrict__ out, long total) {
  long i = (long)blockIdx.x * blockDim.x + threadIdx.x;
  if (i >= total) return;
  int c = (int)(i & (H - 1));
  out[i] = gamma_n[c] * (hacc[i] - nparams[c]) * nparams[128 + c] + beta_n[c] + x_in[i];
}

extern "C" void kernel_launch(void* const* d_in, const int* in_sizes, int n_in,
                              void* d_out, int out_size, void* d_ws, size_t ws_size,
                              hipStream_t stream) {
  const float* x_in    = (const float*)d_in[0];
  const int*   src     = (const int*)d_in[1];
  const int*   dst     = (const int*)d_in[2];
  const float* W_f     = (const float*)d_in[3];
  /* d_in[4] = b_f    : cancels inside batch-norm */
  const float* gamma_f = (const float*)d_in[5];
  const float* beta_f  = (const float*)d_in[6];
  const float* Wg      = (const float*)d_in[7];
  /* d_in[8] = bg     : cancels inside batch-norm */
  const float* gamma_g = (const float*)d_in[9];
  const float* beta_g  = (const float*)d_in[10];
  const float* gamma_n = (const float*)d_in[11];
  const float* beta_n  = (const float*)d_in[12];
  float*       out     = (float*)d_out;

  int N = in_sizes[0] / H;               // 50000
  int E = in_sizes[1];                   // 800000

  // workspace layout (floats)
  float* ws = (float*)d_ws;
  long off = 0;
  float* W2      = ws + off; off += (long)H * PSTR;   // fused weights
  float* P       = ws + off; off += (long)N * PSTR;   // node projections + scores
  float* exp_arr = ws + off; off += E;                // per-edge exp
  float* sum_exp = ws + off; off += N;                // softmax denominators
  float* hacc    = ws + off; off += (long)N * H;      // scatter accumulator
  float* estats  = ws + off; off += 258;              // edge sums/sumsq
  float* eparams = ws + off; off += 258;              // edge mean/invstd
  float* nstats  = ws + off; off += 256;              // node sums/sumsq
  float* nparams = ws + off; off += 256;              // node mean/invstd

  // zero sum_exp .. nparams (contiguous span)
  long zspan = (long)N + (long)N * H + 258 + 258 + 256 + 256;
  k_zero<<<2048, 256, 0, stream>>>(sum_exp, zspan);

  k_build_w2<<<(H * PSTR + 255) / 256, 256, 0, stream>>>(W_f, Wg, W2);

  // WMMA GEMM: grid (N/16 m-tiles, 5 groups of 4 n-tiles), 4 waves/block
  dim3 ggrid(N / 16, (PSTR / 16 + 3) / 4);
  k_gemm_wmma<<<ggrid, 128, 0, stream>>>(x_in, W2, P);

  k_edge_stats<<<512, 256, 0, stream>>>(P, src, dst, estats, E);
  k_fin_edge<<<1, 160, 0, stream>>>(estats, eparams, E);
  k_edge_exp<<<(E + 255) / 256, 256, 0, stream>>>(P, src, dst, eparams, gamma_f, beta_f,
                                                  exp_arr, sum_exp, E);
  k_edge_scatter<<<1024, 256, 0, stream>>>(P, src, dst, eparams, gamma_g, beta_g,
                                           exp_arr, sum_exp, hacc, E);
  k_node_stats<<<200, 128, 0, stream>>>(hacc, nstats, N);
  k_fin_node<<<1, 128, 0, stream>>>(nstats, nparams, N);
  k_out<<<((long)N * H + 255) / 256, 256, 0, stream>>>(hacc, nparams, gamma_n, beta_n,
                                                       x_in, out, (long)N * H);
}